// Down_11802570130432
// MI455X (gfx1250) — compile-verified
//
#include <hip/hip_runtime.h>
#include <hip/hip_bf16.h>

typedef __attribute__((ext_vector_type(2))) float v2f;
typedef __attribute__((ext_vector_type(8))) float v8f;

#define V_VERTS 786432
#define E_EDGES 6291456
#define KCHEB 5
#define FOUT 32

// ---------------------------------------------------------------------------
// zero a float buffer
__global__ void zero_kernel(float* __restrict__ p, int n) {
    int i = blockIdx.x * blockDim.x + threadIdx.x;
    if (i < n) p[i] = 0.0f;
}

// ---------------------------------------------------------------------------
// SpMM (COO): acc[row, :] += val * z[col, :]   (one thread per edge)
__global__ void spmm_atomic(const int* __restrict__ rows,
                            const int* __restrict__ cols,
                            const float* __restrict__ vals,
                            const float* __restrict__ z,
                            float* __restrict__ acc,
                            int nE, int F) {
    int e = blockIdx.x * blockDim.x + threadIdx.x;
    if (e >= nE) return;
    int r = rows[e];
    int c = cols[e];
    float v = vals[e];
    const float* zp = z + (long)c * F;
    float* ap = acc + (long)r * F;
    __builtin_prefetch(zp, 0, 0);   // global_prefetch_b8
    #pragma unroll 4
    for (int f = 0; f < F; ++f) {
        atomicAdd(&ap[f], v * zp[f]);
    }
}

// ---------------------------------------------------------------------------
// Chebyshev recurrence combine: x2 = 2*x2 - x0  (elementwise)
__global__ void cheb_combine(float* __restrict__ x2,
                             const float* __restrict__ x0, int n) {
    int i = blockIdx.x * blockDim.x + threadIdx.x;
    if (i < n) x2[i] = 2.0f * x2[i] - x0[i];
}

// ---------------------------------------------------------------------------
// Fused Chebyshev GEMM + bias + ELU using V_WMMA_F32_16X16X4_F32.
// out[v, o] = elu( sum_{f,k} xk[v,f] * w[f*K + k, o] + bias[o] )
// One wave (32 lanes) owns a 16-vertex M tile; two 16-wide N tiles cover FOUT=32.
template <int FIN>
__global__ __launch_bounds__(256)
void cheb_gemm_elu(const float* __restrict__ x0, const float* __restrict__ x1,
                   const float* __restrict__ x2, const float* __restrict__ x3,
                   const float* __restrict__ x4,
                   const float* __restrict__ w,     // [FIN*KCHEB, FOUT]
                   const float* __restrict__ bias,  // [FOUT]
                   float* __restrict__ out)         // [V, FOUT]
{
    const int lane  = threadIdx.x & 31;
    const int wave  = threadIdx.x >> 5;
    const int tile  = blockIdx.x * 8 + wave;   // 16 vertices per wave
    const int vbase = tile * 16;

    const int m  = lane & 15;     // A row / B col / C col within tile
    const int hi = lane >> 4;     // 0: K 0..1, rows 0..7 | 1: K 2..3, rows 8..15
    const int k0 = hi * 2;

    const float* xs[KCHEB] = {x0, x1, x2, x3, x4};

    v8f c0 = {};
    v8f c1 = {};

    #pragma unroll
    for (int k = 0; k < KCHEB; ++k) {
        const float* __restrict__ X = xs[k];
        #pragma unroll
        for (int f4 = 0; f4 < FIN / 4; ++f4) {
            // A fragment: 16x4 fp32 tile of Chebyshev state k
            const int fb = f4 * 4 + k0;
            v2f a;
            a.x = X[(long)(vbase + m) * FIN + fb];
            a.y = X[(long)(vbase + m) * FIN + fb + 1];

            // B fragments: rows (f*K + k) of w, N tiles [0,16) and [16,32)
            const int wr0 = (f4 * 4 + k0) * KCHEB + k;
            const int wr1 = (f4 * 4 + k0 + 1) * KCHEB + k;
            v2f b0, b1;
            b0.x = w[wr0 * FOUT + m];
            b0.y = w[wr1 * FOUT + m];
            b1.x = w[wr0 * FOUT + 16 + m];
            b1.y = w[wr1 * FOUT + 16 + m];

            c0 = __builtin_amdgcn_wmma_f32_16x16x4_f32(false, a, false, b0,
                                                       (short)0, c0, false, false);
            c1 = __builtin_amdgcn_wmma_f32_16x16x4_f32(false, a, false, b1,
                                                       (short)0, c1, false, false);
        }
    }

    // Epilogue: C/D layout -> VGPR i holds row (i + 8*hi), col = lane&15.
    const float bia0 = bias[m];
    const float bia1 = bias[16 + m];
    #pragma unroll
    for (int i = 0; i < 8; ++i) {
        const int row = vbase + i + hi * 8;
        float v0 = c0[i] + bia0;
        float v1 = c1[i] + bia1;
        v0 = (v0 > 0.0f) ? v0 : (__expf(v0) - 1.0f);
        v1 = (v1 > 0.0f) ? v1 : (__expf(v1) - 1.0f);
        out[(long)row * FOUT + m]      = v0;
        out[(long)row * FOUT + 16 + m] = v1;
    }
}

// ---------------------------------------------------------------------------
// HealpixMaxPool: pooled[v4, f] = max over 4 consecutive vertices
__global__ void pool_kernel(const float* __restrict__ skip,
                            float* __restrict__ pooled, int n) {
    int i = blockIdx.x * blockDim.x + threadIdx.x;
    if (i >= n) return;
    int f  = i & 31;
    int v4 = i >> 5;
    const float* s = skip + (long)v4 * 4 * FOUT + f;
    float mx = s[0];
    mx = fmaxf(mx, s[FOUT]);
    mx = fmaxf(mx, s[2 * FOUT]);
    mx = fmaxf(mx, s[3 * FOUT]);
    pooled[i] = mx;
}

// ---------------------------------------------------------------------------
extern "C" void kernel_launch(void* const* d_in, const int* in_sizes, int n_in,
                              void* d_out, int out_size, void* d_ws, size_t ws_size,
                              hipStream_t stream) {
    const float* x    = (const float*)d_in[0];   // [V, 16]
    const int*   rows = (const int*)d_in[1];     // [E]
    const int*   cols = (const int*)d_in[2];     // [E]
    const float* vals = (const float*)d_in[3];   // [E]
    const float* w1   = (const float*)d_in[4];   // [80, 32]
    const float* b1   = (const float*)d_in[5];   // [32]
    const float* w2   = (const float*)d_in[6];   // [160, 32]
    const float* b2   = (const float*)d_in[7];   // [32]

    float* skip   = (float*)d_out;               // [V, 32]
    float* pooled = skip + (long)V_VERTS * FOUT; // [V/4, 32]

    // 5 scratch buffers of V*32 floats each
    float* buf[5];
    for (int i = 0; i < 5; ++i)
        buf[i] = (float*)d_ws + (long)i * V_VERTS * FOUT;

    const int TB = 256;
    const int eGrid = (E_EDGES + TB - 1) / TB;

    auto zero = [&](float* p, long n) {
        zero_kernel<<<(int)((n + TB - 1) / TB), TB, 0, stream>>>(p, (int)n);
    };
    auto spmm = [&](float* acc, const float* z, int F) {
        zero(acc, (long)V_VERTS * F);
        spmm_atomic<<<eGrid, TB, 0, stream>>>(rows, cols, vals, z, acc, E_EDGES, F);
    };
    auto combine = [&](float* x2, const float* x0, int F) {
        long n = (long)V_VERTS * F;
        cheb_combine<<<(int)((n + TB - 1) / TB), TB, 0, stream>>>(x2, x0, (int)n);
    };

    // ---------------- Layer 1 (Fin = 16) ----------------
    // x0 = input x; x1..x4 stored compact [V,16] in buf1..buf4
    spmm(buf[1], x, 16);                      // x1 = L x0
    spmm(buf[2], buf[1], 16);                 // x2 = 2 L x1 - x0
    combine(buf[2], x, 16);
    spmm(buf[3], buf[2], 16);                 // x3 = 2 L x2 - x1
    combine(buf[3], buf[1], 16);
    spmm(buf[4], buf[3], 16);                 // x4 = 2 L x3 - x2
    combine(buf[4], buf[2], 16);

    // H1 = elu( [x0..x4] @ w1 + b1 )  -> buf0  [V, 32]
    cheb_gemm_elu<16><<<V_VERTS / 128, 256, 0, stream>>>(
        x, buf[1], buf[2], buf[3], buf[4], w1, b1, buf[0]);

    // ---------------- Layer 2 (Fin = 32) ----------------
    spmm(buf[1], buf[0], 32);                 // x1 = L H1
    spmm(buf[2], buf[1], 32);                 // x2 = 2 L x1 - x0
    combine(buf[2], buf[0], 32);
    spmm(buf[3], buf[2], 32);                 // x3
    combine(buf[3], buf[1], 32);
    spmm(buf[4], buf[3], 32);                 // x4
    combine(buf[4], buf[2], 32);

    // skip = elu( [H1, x1..x4] @ w2 + b2 )  -> d_out
    cheb_gemm_elu<32><<<V_VERTS / 128, 256, 0, stream>>>(
        buf[0], buf[1], buf[2], buf[3], buf[4], w2, b2, skip);

    // ---------------- Pool ----------------
    const long np = (long)(V_VERTS / 4) * FOUT;
    pool_kernel<<<(int)((np + TB - 1) / TB), TB, 0, stream>>>(skip, pooled, (int)np);
}